// YChannelEnhancementLoss_13434657702751
// MI455X (gfx1250) — compile-verified
//
#include <hip/hip_runtime.h>
#include <math.h>

// Problem constants (fixed by the reference's setup_inputs)
#define BATCH 32
#define IMG_H 1024
#define IMG_W 1024
#define NUM_BINS 256
// pass1 launch geometry
#define THREADS 256
#define BLOCKS_X 256                          // blocks per image
#define GROUPS_PER_IMAGE (IMG_H * (IMG_W / 4)) // float4 groups: 262144
#define GROUPS_PER_BLOCK (GROUPS_PER_IMAGE / BLOCKS_X) // 1024
#define ITERS (GROUPS_PER_BLOCK / THREADS)    // 4
#define NPARTIAL (BATCH * BLOCKS_X)           // 8192 blocks total

typedef __attribute__((ext_vector_type(2))) float v2f;
typedef __attribute__((ext_vector_type(8))) float v8f;

// Exact, deterministic 32-lane sum using V_WMMA_F32_16X16X4_F32.
// A: VGPR0 = v (covers A[m,0] from lanes 0-15, A[m,2] from lanes 16-31),
//    VGPR1 = 0 (K=1,3). B = all ones (4x16). C = 0.
// => D[i][j] = v[i] + v[i+16] for every column j.
// Summing the 8 D VGPRs per lane gives sum(v[0..7]+v[16..23]) in lanes 0-15
// and sum(v[8..15]+v[24..31]) in lanes 16-31; one xor-16 shuffle completes it.
__device__ __forceinline__ float wave_sum32(float v) {
#if __has_builtin(__builtin_amdgcn_wmma_f32_16x16x4_f32)
  v2f a; a[0] = v;    a[1] = 0.0f;
  v2f b; b[0] = 1.0f; b[1] = 1.0f;
  v8f c = {};
  c = __builtin_amdgcn_wmma_f32_16x16x4_f32(false, a, false, b, (short)0, c,
                                            false, false);
  float s = c[0] + c[1] + c[2] + c[3] + c[4] + c[5] + c[6] + c[7];
  s += __shfl_xor(s, 16, 32);
  return s;
#else
  float s = v;
  for (int m = 16; m >= 1; m >>= 1) s += __shfl_xor(s, m, 32);
  return s;
#endif
}

// Fused streaming pass: l1, |laplacian| for both images, dx/dy TV terms for
// enhanced, and per-batch 256-bin histograms for both images.
__global__ void __launch_bounds__(THREADS)
ychannel_pass1(const float* __restrict__ e, const float* __restrict__ o,
               unsigned* __restrict__ hist_e, unsigned* __restrict__ hist_o,
               float* __restrict__ partials) {
  __shared__ unsigned lhe[NUM_BINS];
  __shared__ unsigned lho[NUM_BINS];
  __shared__ float wsum[THREADS / 32][5];

  const int t  = threadIdx.x;
  const int b  = blockIdx.y;   // batch image
  const int bx = blockIdx.x;   // chunk within image

  lhe[t] = 0u;
  lho[t] = 0u;
  __syncthreads();

  const size_t base = (size_t)b * (IMG_H * (size_t)IMG_W);
  const float* eb = e + base;
  const float* ob = o + base;

  float a_l1 = 0.f, a_ce = 0.f, a_co = 0.f, a_dx = 0.f, a_dy = 0.f;

#pragma unroll
  for (int k = 0; k < ITERS; ++k) {
    const int g  = bx * GROUPS_PER_BLOCK + k * THREADS + t; // float4 group id
    const int i  = g >> 8;           // row (W/4 == 256 groups per row)
    const int j0 = (g & 255) << 2;   // starting column of the 4-pixel group
    const size_t row = (size_t)i * IMG_W;

    if (k + 1 < ITERS) {
      __builtin_prefetch(eb + row + THREADS * 4, 0, 1);
      __builtin_prefetch(ob + row + THREADS * 4, 0, 1);
    }

    const float4 ec4 = *(const float4*)(eb + row + j0);
    const float4 oc4 = *(const float4*)(ob + row + j0);
    float4 eu4 = make_float4(0.f, 0.f, 0.f, 0.f), ed4 = eu4;
    float4 ou4 = eu4, od4 = eu4;
    if (i > 0) {
      eu4 = *(const float4*)(eb + row - IMG_W + j0);
      ou4 = *(const float4*)(ob + row - IMG_W + j0);
    }
    if (i < IMG_H - 1) {
      ed4 = *(const float4*)(eb + row + IMG_W + j0);
      od4 = *(const float4*)(ob + row + IMG_W + j0);
    }
    float eL = 0.f, eR = 0.f, oL = 0.f, oR = 0.f;
    if (j0 > 0)          { eL = eb[row + j0 - 1]; oL = ob[row + j0 - 1]; }
    if (j0 + 4 < IMG_W)  { eR = eb[row + j0 + 4]; oR = ob[row + j0 + 4]; }

    const float ec[4] = {ec4.x, ec4.y, ec4.z, ec4.w};
    const float oc[4] = {oc4.x, oc4.y, oc4.z, oc4.w};
    const float eu[4] = {eu4.x, eu4.y, eu4.z, eu4.w};
    const float ed[4] = {ed4.x, ed4.y, ed4.z, ed4.w};
    const float ou[4] = {ou4.x, ou4.y, ou4.z, ou4.w};
    const float od[4] = {od4.x, od4.y, od4.z, od4.w};
    const float el[4] = {eL, ec[0], ec[1], ec[2]};
    const float er[4] = {ec[1], ec[2], ec[3], eR};
    const float ol[4] = {oL, oc[0], oc[1], oc[2]};
    const float orr[4] = {oc[1], oc[2], oc[3], oR};

#pragma unroll
    for (int q = 0; q < 4; ++q) {
      a_l1 += fabsf(ec[q] - oc[q]);
      a_ce += fabsf(eu[q] + ed[q] + el[q] + er[q] - 4.0f * ec[q]);
      a_co += fabsf(ou[q] + od[q] + ol[q] + orr[q] - 4.0f * oc[q]);
      // Histogram bucket: exact searchsorted('left') - 1 (x*256 is exact).
      const int be_ = (int)ceilf(ec[q] * 256.0f) - 1;
      if (be_ >= 0 && be_ < NUM_BINS) atomicAdd(&lhe[be_], 1u);
      const int bo_ = (int)ceilf(oc[q] * 256.0f) - 1;
      if (bo_ >= 0 && bo_ < NUM_BINS) atomicAdd(&lho[bo_], 1u);
    }
    if (i < IMG_H - 1) {
#pragma unroll
      for (int q = 0; q < 4; ++q) a_dx += fabsf(ed[q] - ec[q]);
    }
    a_dy += fabsf(ec[1] - ec[0]) + fabsf(ec[2] - ec[1]) + fabsf(ec[3] - ec[2]);
    if (j0 + 4 < IMG_W) a_dy += fabsf(eR - ec[3]);
  }

  __syncthreads();
  // Merge per-block histograms into per-batch global histograms (int atomics:
  // order-independent => deterministic).
  atomicAdd(&hist_e[b * NUM_BINS + t], lhe[t]);
  atomicAdd(&hist_o[b * NUM_BINS + t], lho[t]);

  // Deterministic block reduction: WMMA wave sums, then serial 8-way combine.
  const float s0 = wave_sum32(a_l1);
  const float s1 = wave_sum32(a_ce);
  const float s2 = wave_sum32(a_co);
  const float s3 = wave_sum32(a_dx);
  const float s4 = wave_sum32(a_dy);
  const int wv = t >> 5, ln = t & 31;
  if (ln == 0) {
    wsum[wv][0] = s0; wsum[wv][1] = s1; wsum[wv][2] = s2;
    wsum[wv][3] = s3; wsum[wv][4] = s4;
  }
  __syncthreads();
  if (t == 0) {
    float r[5] = {0.f, 0.f, 0.f, 0.f, 0.f};
    for (int w = 0; w < THREADS / 32; ++w)
      for (int q = 0; q < 5; ++q) r[q] += wsum[w][q];
    const size_t pb = ((size_t)b * gridDim.x + bx) * 5;
    for (int q = 0; q < 5; ++q) partials[pb + q] = r[q];
  }
}

// Single-block finalize: deterministic double-precision tree reductions.
__global__ void __launch_bounds__(THREADS)
ychannel_finalize(const unsigned* __restrict__ hist_e,
                  const unsigned* __restrict__ hist_o,
                  const float* __restrict__ partials, int P,
                  float* __restrict__ out) {
  __shared__ double red[THREADS];
  __shared__ double finals[5];
  __shared__ double Se[BATCH], So[BATCH];
  const int t = threadIdx.x;

  for (int q = 0; q < 5; ++q) {
    double acc = 0.0;
    for (int i = t; i < P; i += THREADS) acc += (double)partials[(size_t)i * 5 + q];
    red[t] = acc;
    __syncthreads();
    for (int s = THREADS / 2; s > 0; s >>= 1) {
      if (t < s) red[t] += red[t + s];
      __syncthreads();
    }
    if (t == 0) finals[q] = red[0];
    __syncthreads();
  }

  if (t < BATCH) {
    double se = 0.0, so = 0.0;
    for (int k = 0; k < NUM_BINS; ++k) {
      se += (double)hist_e[t * NUM_BINS + k];
      so += (double)hist_o[t * NUM_BINS + k];
    }
    Se[t] = se; So[t] = so;
  }
  __syncthreads();

  double hacc = 0.0; // thread t owns bin t across all batches
  for (int b = 0; b < BATCH; ++b) {
    const double hn_e = ((double)hist_e[b * NUM_BINS + t] + 1e-6) / (Se[b] + 1e-6);
    const double hn_o = ((double)hist_o[b * NUM_BINS + t] + 1e-6) / (So[b] + 1e-6);
    const double d = hn_e - hn_o;
    hacc += d * d;
  }
  red[t] = hacc;
  __syncthreads();
  for (int s = THREADS / 2; s > 0; s >>= 1) {
    if (t < s) red[t] += red[t + s];
    __syncthreads();
  }

  if (t == 0) {
    const double N = (double)BATCH * IMG_H * IMG_W;
    const double hist_loss = red[0] / ((double)BATCH * NUM_BINS) / (double)NUM_BINS;
    const double l1 = finals[0] / N;
    const double c_e = finals[1] / N;
    const double c_o = finals[2] / N;
    const double cont = fabs(c_e - c_o) / (c_o + 1e-6);
    const double smooth = finals[3] / ((double)BATCH * (IMG_H - 1) * IMG_W) +
                          finals[4] / ((double)BATCH * IMG_H * (IMG_W - 1));
    out[0] = (float)(l1 + 0.1 * hist_loss + 0.1 * cont + 0.01 * smooth);
  }
}

extern "C" void kernel_launch(void* const* d_in, const int* in_sizes, int n_in,
                              void* d_out, int out_size, void* d_ws, size_t ws_size,
                              hipStream_t stream) {
  const float* e = (const float*)d_in[0];
  const float* o = (const float*)d_in[1];

  // Workspace layout:
  //   [0, 32KB)      hist_e : u32 [BATCH][256]
  //   [32KB, 64KB)   hist_o : u32 [BATCH][256]
  //   [64KB, ...)    partials: float [NPARTIAL][5]
  unsigned* hist_e = (unsigned*)d_ws;
  unsigned* hist_o = hist_e + BATCH * NUM_BINS;
  float* partials = (float*)((char*)d_ws + 2u * BATCH * NUM_BINS * sizeof(unsigned));

  hipMemsetAsync(d_ws, 0, 2u * BATCH * NUM_BINS * sizeof(unsigned), stream);

  dim3 grid(BLOCKS_X, BATCH);
  ychannel_pass1<<<grid, THREADS, 0, stream>>>(e, o, hist_e, hist_o, partials);
  ychannel_finalize<<<1, THREADS, 0, stream>>>(hist_e, hist_o, partials,
                                               NPARTIAL, (float*)d_out);
}